// GaitGraph_62122406969795
// MI455X (gfx1250) — compile-verified
//
#include <hip/hip_runtime.h>
#include <hip/hip_bf16.h>

typedef __attribute__((ext_vector_type(2))) float v2f;
typedef __attribute__((ext_vector_type(8))) float v8f;

#define H64 64

// ---------------- init: zero accumulators, deg <- 1.0 (self loop) ----------------
__global__ void init_kernel(float* __restrict__ stats, float* __restrict__ deg,
                            float* __restrict__ bufQ, float* __restrict__ pooled,
                            float* __restrict__ cnt, int N, int G) {
    long long i = blockIdx.x * (long long)blockDim.x + threadIdx.x;
    if (i < 4)                  stats[i]  = 0.0f;
    if (i < N)                  deg[i]    = 1.0f;     // self loop contributes 1 to degree
    if (i < (long long)N * H64) bufQ[i]   = 0.0f;
    if (i < (long long)G * H64) pooled[i] = 0.0f;
    if (i < G)                  cnt[i]    = 0.0f;
}

// ---------------- feature stats: sum and sumsq of nan_to_num(x), 2 features ------
__global__ void stats_kernel(const float* __restrict__ x, float* __restrict__ stats, int N) {
    int i = blockIdx.x * blockDim.x + threadIdx.x;
    float v0 = 0.0f, v1 = 0.0f;
    if (i < N) {
        v0 = x[2 * i];
        v1 = x[2 * i + 1];
        if (!__builtin_isfinite(v0)) v0 = 0.0f;
        if (!__builtin_isfinite(v1)) v1 = 0.0f;
    }
    float s0 = v0, s1 = v1, q0 = v0 * v0, q1 = v1 * v1;
    for (int m = 16; m > 0; m >>= 1) {  // wave32 reduce
        s0 += __shfl_xor(s0, m, 32);
        s1 += __shfl_xor(s1, m, 32);
        q0 += __shfl_xor(q0, m, 32);
        q1 += __shfl_xor(q1, m, 32);
    }
    if ((threadIdx.x & 31) == 0) {
        atomicAdd(&stats[0], s0);
        atomicAdd(&stats[1], s1);
        atomicAdd(&stats[2], q0);
        atomicAdd(&stats[3], q1);
    }
}

// ---------------- degree accumulation over edge targets --------------------------
__global__ void deg_kernel(const int* __restrict__ dst, float* __restrict__ deg, long long E) {
    long long e = blockIdx.x * (long long)blockDim.x + threadIdx.x;
    if (e < E) atomicAdd(&deg[dst[e]], 1.0f);
}

__global__ void dinv_kernel(float* __restrict__ deg, int N) {
    int i = blockIdx.x * blockDim.x + threadIdx.x;
    if (i < N) deg[i] = rsqrtf(deg[i]);   // deg >= 1 always (self loop)
}

// ---------------- fused standardize + x @ W1 (IN=2) ------------------------------
__global__ void l1_transform(const float* __restrict__ x, const float* __restrict__ W1,
                             const float* __restrict__ stats, float* __restrict__ outP, int N) {
    long long i = blockIdx.x * (long long)blockDim.x + threadIdx.x;
    if (i >= (long long)N * H64) return;
    int n = (int)(i >> 6), j = (int)(i & 63);
    float acc = 0.0f;
    #pragma unroll
    for (int f = 0; f < 2; ++f) {
        float mu  = stats[f] / (float)N;
        float var = (stats[2 + f] - (float)N * mu * mu) / (float)(N - 1);
        var = var > 0.0f ? var : 0.0f;
        float inv = 1.0f / (sqrtf(var) + 1e-6f);
        float v = x[2 * n + f];
        if (!__builtin_isfinite(v)) v = 0.0f;
        acc += (v - mu) * inv * W1[f * H64 + j];
    }
    outP[i] = acc;
}

// ---------------- edge scatter: one wave per edge, 2 features per lane ------------
__global__ void scatter_kernel(const float* __restrict__ hsrc, float* __restrict__ hacc,
                               const int* __restrict__ src, const int* __restrict__ dst,
                               const float* __restrict__ dinv, long long E) {
    long long t = blockIdx.x * (long long)blockDim.x + threadIdx.x;
    long long e = t >> 5;
    int f = (int)(t & 31);
    if (e >= E) return;
    int s = src[e], d = dst[e];
    float nrm = dinv[s] * dinv[d];
    atomicAdd(&hacc[(long long)d * H64 + f],      hsrc[(long long)s * H64 + f]      * nrm);
    atomicAdd(&hacc[(long long)d * H64 + f + 32], hsrc[(long long)s * H64 + f + 32] * nrm);
}

// ---------------- finalize layer1: edge sums + self loop + bias, ReLU (in place) --
__global__ void finalize1(float* __restrict__ P, const float* __restrict__ Q,
                          const float* __restrict__ dinv, const float* __restrict__ b, int N) {
    long long i = blockIdx.x * (long long)blockDim.x + threadIdx.x;
    if (i >= (long long)N * H64) return;
    int n = (int)(i >> 6), j = (int)(i & 63);
    float di = dinv[n];
    float v = Q[i] + P[i] * di * di + b[j];
    P[i] = v > 0.0f ? v : 0.0f;
}

// ---------------- WMMA fp32 GEMM: D[N,64] = A[N,64] @ W[64,64] --------------------
// One wave per 16x16 output tile; K-loop of 16 x V_WMMA_F32_16X16X4_F32.
// 32-bit A 16x4 layout: lanes 0-15 -> M=lane, v0=K0 v1=K1; lanes 16-31 -> v0=K2 v1=K3.
// 32-bit D 16x16 layout: VGPR r -> M = r + (lane>=16 ? 8 : 0), Nc = lane&15.
__global__ __launch_bounds__(128)
void gemm64_wmma(const float* __restrict__ A, const float* __restrict__ W,
                 float* __restrict__ D, int N) {
    int lane    = threadIdx.x;        // 0..31
    int colBase = threadIdx.y * 16;   // 4 waves -> 4 column tiles of 64
    int rowBase = blockIdx.x * 16;
    int half    = lane >> 4;          // 0: K+0/K+1, 1: K+2/K+3
    int l       = lane & 15;
    int arow    = rowBase + l;
    if (arow >= N) arow = N - 1;      // clamp (EXEC must stay all-ones for WMMA)
    v8f acc = {};
    #pragma unroll
    for (int k = 0; k < H64; k += 4) {
        int ka = k + half * 2;
        v2f a, b;
        a.x = A[(long long)arow * H64 + ka];
        a.y = A[(long long)arow * H64 + ka + 1];
        b.x = W[(long long)ka * H64 + colBase + l];
        b.y = W[(long long)(ka + 1) * H64 + colBase + l];
        acc = __builtin_amdgcn_wmma_f32_16x16x4_f32(false, a, false, b,
                                                    (short)0, acc, false, false);
    }
    #pragma unroll
    for (int r = 0; r < 8; ++r) {
        int m = rowBase + r + half * 8;
        if (m < N) D[(long long)m * H64 + colBase + l] = acc[r];
    }
}

// ---------------- zero a [N,64] buffer --------------------------------------------
__global__ void zero_kernel(float* __restrict__ p, long long n) {
    long long i = blockIdx.x * (long long)blockDim.x + threadIdx.x;
    if (i < n) p[i] = 0.0f;
}

// ---------------- finalize layer2 + pooled accumulation ---------------------------
__global__ void finalize2_pool(const float* __restrict__ P, const float* __restrict__ Q,
                               const float* __restrict__ dinv, const float* __restrict__ b,
                               const int* __restrict__ batch, float* __restrict__ pooled, int N) {
    long long i = blockIdx.x * (long long)blockDim.x + threadIdx.x;
    if (i >= (long long)N * H64) return;
    int n = (int)(i >> 6), j = (int)(i & 63);
    float di = dinv[n];
    float v = P[i] + Q[i] * di * di + b[j];
    v = v > 0.0f ? v : 0.0f;
    atomicAdd(&pooled[(long long)batch[n] * H64 + j], v);
}

__global__ void counts_kernel(const int* __restrict__ batch, float* __restrict__ cnt, int N) {
    int i = blockIdx.x * blockDim.x + threadIdx.x;
    if (i < N) atomicAdd(&cnt[batch[i]], 1.0f);
}

// ---------------- classifier: out[G,C] = (pooled/cnt) @ Wc + bc -------------------
__global__ void classifier_kernel(const float* __restrict__ pooled, const float* __restrict__ cnt,
                                  const float* __restrict__ Wc, const float* __restrict__ bc,
                                  float* __restrict__ out, int G, int C) {
    int i = blockIdx.x * blockDim.x + threadIdx.x;
    if (i >= G * C) return;
    int g = i / C, c = i % C;
    float inv = 1.0f / fmaxf(cnt[g], 1.0f);
    float acc = bc[c];
    #pragma unroll 8
    for (int j = 0; j < H64; ++j)
        acc += pooled[(long long)g * H64 + j] * inv * Wc[j * C + c];
    out[i] = acc;
}

extern "C" void kernel_launch(void* const* d_in, const int* in_sizes, int n_in,
                              void* d_out, int out_size, void* d_ws, size_t ws_size,
                              hipStream_t stream) {
    const float* x    = (const float*)d_in[0];
    const float* W1   = (const float*)d_in[1];
    const float* b1   = (const float*)d_in[2];
    const float* W2   = (const float*)d_in[3];
    const float* b2   = (const float*)d_in[4];
    const float* Wc   = (const float*)d_in[5];
    const float* bc   = (const float*)d_in[6];
    const int*   eidx = (const int*)d_in[7];
    const int*   batch= (const int*)d_in[8];

    const int       N  = in_sizes[8];          // 50000
    const long long E  = in_sizes[7] / 2;      // 800000
    const int       H  = in_sizes[2];          // 64
    const int       C  = in_sizes[5] / H;      // 3
    const int       G  = out_size / C;         // 2048
    const int* src = eidx;
    const int* dst = eidx + E;

    // workspace layout (floats)
    float* ws = (float*)d_ws;
    size_t o = 0;
    float* stats  = ws + o; o += 64;
    float* dinv   = ws + o; o += ((size_t)N + 63) & ~(size_t)63;
    float* bufP   = ws + o; o += (size_t)N * H64;
    float* bufQ   = ws + o; o += (size_t)N * H64;
    float* pooled = ws + o; o += (size_t)G * H64;
    float* cnt    = ws + o; o += (size_t)G;
    (void)ws_size; (void)n_in;

    const long long NH = (long long)N * H64;
    const int TB = 256;
    dim3 gNH((unsigned)((NH + TB - 1) / TB));
    dim3 gN((unsigned)((N + TB - 1) / TB));
    dim3 gE((unsigned)((E + TB - 1) / TB));
    dim3 gE32((unsigned)((E * 32 + TB - 1) / TB));

    init_kernel<<<gNH, TB, 0, stream>>>(stats, dinv, bufQ, pooled, cnt, N, G);
    stats_kernel<<<gN, TB, 0, stream>>>(x, stats, N);
    deg_kernel<<<gE, TB, 0, stream>>>(dst, dinv, E);
    dinv_kernel<<<gN, TB, 0, stream>>>(dinv, N);

    // layer 1
    l1_transform<<<gNH, TB, 0, stream>>>(x, W1, stats, bufP, N);
    scatter_kernel<<<gE32, TB, 0, stream>>>(bufP, bufQ, src, dst, dinv, E);
    finalize1<<<gNH, TB, 0, stream>>>(bufP, bufQ, dinv, b1, N);

    // layer 2: WMMA GEMM bufQ = bufP @ W2
    dim3 gGemm((unsigned)((N + 15) / 16));
    dim3 bGemm(32, 4);
    gemm64_wmma<<<gGemm, bGemm, 0, stream>>>(bufP, W2, bufQ, N);
    zero_kernel<<<gNH, TB, 0, stream>>>(bufP, NH);
    scatter_kernel<<<gE32, TB, 0, stream>>>(bufQ, bufP, src, dst, dinv, E);
    finalize2_pool<<<gNH, TB, 0, stream>>>(bufP, bufQ, dinv, b2, batch, pooled, N);

    // pooling + classifier
    counts_kernel<<<gN, TB, 0, stream>>>(batch, cnt, N);
    dim3 gCls((unsigned)((G * C + TB - 1) / TB));
    classifier_kernel<<<gCls, TB, 0, stream>>>(pooled, cnt, Wc, bc, (float*)d_out, G, C);
}